// PostNormAttentionBlock_17867063952096
// MI455X (gfx1250) — compile-verified
//
#include <hip/hip_runtime.h>
#include <math.h>

typedef __attribute__((ext_vector_type(2))) float v2f;
typedef __attribute__((ext_vector_type(8))) float v8f;

// D = A(16x4,f32) * B(4x16,f32) + C(16x16,f32)   -- CDNA5 V_WMMA_F32_16X16X4_F32
__device__ __forceinline__ v8f wmma_f32(v2f a, v2f b, v8f c) {
  return __builtin_amdgcn_wmma_f32_16x16x4_f32(false, a, false, b, (short)0, c,
                                               false, false);
}

__device__ __forceinline__ float hmax16(float v) {
  v = fmaxf(v, __shfl_xor(v, 1));
  v = fmaxf(v, __shfl_xor(v, 2));
  v = fmaxf(v, __shfl_xor(v, 4));
  v = fmaxf(v, __shfl_xor(v, 8));
  return v;
}
__device__ __forceinline__ float hsum16(float v) {
  v += __shfl_xor(v, 1);
  v += __shfl_xor(v, 2);
  v += __shfl_xor(v, 4);
  v += __shfl_xor(v, 8);
  return v;
}

// ---------------------------------------------------------------------------
// Generic fp32 WMMA GEMM:  C[M,N] = act(A[M,K] @ B[K,N] + bias)
// block = 256 threads = 8 waves (4 in M x 2 in N); wave tile = 64x32
// block tile = 256(M) x 64(N).  Requires M%256==0, N%64==0, K%4==0.
// act: 0 = none, 1 = exact GELU
// ---------------------------------------------------------------------------
__global__ __launch_bounds__(256) void gemm_wmma_f32_kernel(
    const float* __restrict__ A, const float* __restrict__ B,
    const float* __restrict__ bias, float* __restrict__ C, int M, int N, int K,
    int lda, int ldb, int ldc, int act) {
  const int lane = threadIdx.x & 31;
  const int wave = threadIdx.x >> 5;
  const int waveM = wave >> 1;  // 0..3
  const int waveN = wave & 1;   // 0..1
  const int m0 = blockIdx.x * 256 + waveM * 64;
  const int n0 = blockIdx.y * 64 + waveN * 32;
  const int lr = lane & 15;
  const int lh = lane >> 4;

  v8f acc[4][2];
#pragma unroll
  for (int i = 0; i < 4; i++)
#pragma unroll
    for (int j = 0; j < 2; j++) acc[i][j] = {};

  for (int k0 = 0; k0 < K; k0 += 4) {
    v2f af[4];
#pragma unroll
    for (int mt = 0; mt < 4; mt++) {
      const float* ap = A + (size_t)(m0 + 16 * mt + lr) * lda + k0 + 2 * lh;
      af[mt] = *(const v2f*)ap;  // contiguous, 8B aligned
    }
    v2f bf[2];
#pragma unroll
    for (int nt = 0; nt < 2; nt++) {
      const float* bp = B + (size_t)(k0 + 2 * lh) * ldb + n0 + 16 * nt + lr;
      bf[nt][0] = bp[0];
      bf[nt][1] = bp[ldb];
    }
#pragma unroll
    for (int mt = 0; mt < 4; mt++)
#pragma unroll
      for (int nt = 0; nt < 2; nt++)
        acc[mt][nt] = wmma_f32(af[mt], bf[nt], acc[mt][nt]);
  }

#pragma unroll
  for (int nt = 0; nt < 2; nt++) {
    const int n = n0 + 16 * nt + lr;
    const float bv = bias[n];
#pragma unroll
    for (int mt = 0; mt < 4; mt++) {
#pragma unroll
      for (int r = 0; r < 8; r++) {
        const int m = m0 + 16 * mt + r + 8 * lh;
        float v = acc[mt][nt][r] + bv;
        if (act == 1) v = 0.5f * v * (1.0f + erff(v * 0.70710678118654752f));
        C[(size_t)m * ldc + n] = v;
      }
    }
  }
}

// ---------------------------------------------------------------------------
// Flash attention, fp32 WMMA, per-head hd=64, scale = 1/8.
// qkv: [S, 3*768]; q at col h*64, k at 768+h*64, v at 1536+h*64.
// out: [S, 768] (heads merged).
// One wave handles (head, 16 query rows). 8 waves/block, grid = H*S/16/8.
// ---------------------------------------------------------------------------
__global__ __launch_bounds__(256) void attn_flash_kernel(
    const float* __restrict__ qkv, float* __restrict__ out, int S) {
  __shared__ float pbuf[8][16 * 17];
  const int lane = threadIdx.x & 31;
  const int wave = threadIdx.x >> 5;
  const int ntiles = S >> 4;
  const int wg = blockIdx.x * 8 + wave;
  const int head = wg / ntiles;
  const int q0 = (wg % ntiles) * 16;
  const int lr = lane & 15;
  const int lh = lane >> 4;
  const int ldq = 3 * 768;

  const float* qp = qkv + head * 64;
  const float* kp = qkv + 768 + head * 64;
  const float* vp = qkv + 1536 + head * 64;

  // Q tile 16x64 fully in registers as 16 A-fragments
  v2f qfrag[16];
#pragma unroll
  for (int ks = 0; ks < 16; ks++)
    qfrag[ks] = *(const v2f*)(qp + (size_t)(q0 + lr) * ldq + 4 * ks + 2 * lh);

  v8f o[4];
#pragma unroll
  for (int t = 0; t < 4; t++) o[t] = {};
  float mrow[8], lrow[8];
#pragma unroll
  for (int r = 0; r < 8; r++) {
    mrow[r] = -INFINITY;
    lrow[r] = 0.0f;
  }

  float* pl = pbuf[wave];

  for (int kv = 0; kv < S; kv += 16) {
    // S_tile = Q @ K^T  (16 WMMA over hd=64)
    v8f s = {};
#pragma unroll
    for (int ks = 0; ks < 16; ks++) {
      v2f kf = *(const v2f*)(kp + (size_t)(kv + lr) * ldq + 4 * ks + 2 * lh);
      s = wmma_f32(qfrag[ks], kf, s);
    }
    // online softmax; C-layout row = r + 8*lh, col = lr
#pragma unroll
    for (int r = 0; r < 8; r++) {
      float sv = s[r] * 0.125f;
      float rmax = hmax16(sv);
      float mnew = fmaxf(mrow[r], rmax);
      float alpha = __expf(mrow[r] - mnew);
      float p = __expf(sv - mnew);
      float rsum = hsum16(p);
      lrow[r] = lrow[r] * alpha + rsum;
      mrow[r] = mnew;
#pragma unroll
      for (int t = 0; t < 4; t++) o[t][r] *= alpha;
      pl[(r + 8 * lh) * 17 + lr] = p;  // stage P for A-frag relayout
    }
    __syncthreads();
    // O += P @ V  (4 k-steps x 4 n-tiles)
#pragma unroll
    for (int ks = 0; ks < 4; ks++) {
      v2f pf;
      pf[0] = pl[lr * 17 + 4 * ks + 2 * lh];
      pf[1] = pl[lr * 17 + 4 * ks + 2 * lh + 1];
#pragma unroll
      for (int t = 0; t < 4; t++) {
        const float* vpp =
            vp + (size_t)(kv + 4 * ks + 2 * lh) * ldq + 16 * t + lr;
        v2f vf;
        vf[0] = vpp[0];
        vf[1] = vpp[ldq];
        o[t] = wmma_f32(pf, vf, o[t]);
      }
    }
    __syncthreads();
  }

#pragma unroll
  for (int r = 0; r < 8; r++) {
    const float inv = 1.0f / lrow[r];
    const int m = q0 + r + 8 * lh;
#pragma unroll
    for (int t = 0; t < 4; t++)
      out[(size_t)m * 768 + head * 64 + 16 * t + lr] = o[t][r] * inv;
  }
}

// ---------------------------------------------------------------------------
// out[row] = LayerNorm(a[row] + b[row]; w, bb), D = 768, one block per row.
// ---------------------------------------------------------------------------
__global__ __launch_bounds__(256) void resid_ln_kernel(
    const float* __restrict__ a, const float* __restrict__ b,
    const float* __restrict__ w, const float* __restrict__ bb,
    float* __restrict__ out) {
  __shared__ float redS[8], redQ[8];
  const int row = blockIdx.x;
  const float* ap = a + (size_t)row * 768;
  const float* bp = b + (size_t)row * 768;
  float v[3], sum = 0.0f, sq = 0.0f;
#pragma unroll
  for (int i = 0; i < 3; i++) {
    const int c = threadIdx.x + 256 * i;
    v[i] = ap[c] + bp[c];
    sum += v[i];
    sq += v[i] * v[i];
  }
#pragma unroll
  for (int m = 1; m < 32; m <<= 1) {
    sum += __shfl_xor(sum, m);
    sq += __shfl_xor(sq, m);
  }
  if ((threadIdx.x & 31) == 0) {
    redS[threadIdx.x >> 5] = sum;
    redQ[threadIdx.x >> 5] = sq;
  }
  __syncthreads();
  float ts = 0.0f, tq = 0.0f;
#pragma unroll
  for (int i = 0; i < 8; i++) {
    ts += redS[i];
    tq += redQ[i];
  }
  const float mean = ts * (1.0f / 768.0f);
  const float var = tq * (1.0f / 768.0f) - mean * mean;
  const float rstd = rsqrtf(var + 1e-5f);
#pragma unroll
  for (int i = 0; i < 3; i++) {
    const int c = threadIdx.x + 256 * i;
    out[(size_t)row * 768 + c] = w[c] * ((v[i] - mean) * rstd) + bb[c];
  }
}

// ---------------------------------------------------------------------------
extern "C" void kernel_launch(void* const* d_in, const int* in_sizes, int n_in,
                              void* d_out, int out_size, void* d_ws,
                              size_t ws_size, hipStream_t stream) {
  const int S = 4096, D = 768, HID = 3072;
  const float* x = (const float*)d_in[0];
  const float* Wqkv = (const float*)d_in[1];
  const float* bqkv = (const float*)d_in[2];
  const float* Wo = (const float*)d_in[3];
  const float* bo = (const float*)d_in[4];
  const float* ln1w = (const float*)d_in[5];
  const float* ln1b = (const float*)d_in[6];
  const float* W1 = (const float*)d_in[7];
  const float* b1 = (const float*)d_in[8];
  const float* W2 = (const float*)d_in[9];
  const float* b2 = (const float*)d_in[10];
  const float* ln2w = (const float*)d_in[11];
  const float* ln2b = (const float*)d_in[12];
  float* out = (float*)d_out;

  float* ws = (float*)d_ws;
  float* qkv = ws;                       // S*3D  = 9,437,184 f
  float* hbuf = ws + (size_t)S * 3 * D;  // S*D
  float* attn = hbuf + (size_t)S * D;    // S*D
  float* x1 = attn + (size_t)S * D;      // S*D
  float* ff1 = ws;                       // S*HID, reuses qkv+hbuf (dead)
  float* ff2 = attn;                     // S*D,   reuses attn (dead)

  dim3 blk(256);

  // 1) qkv = x @ Wqkv + bqkv
  gemm_wmma_f32_kernel<<<dim3(S / 256, (3 * D) / 64), blk, 0, stream>>>(
      x, Wqkv, bqkv, qkv, S, 3 * D, D, D, 3 * D, 3 * D, 0);
  // 2) flash attention -> hbuf [S, D] (heads merged)
  attn_flash_kernel<<<dim3((12 * (S / 16)) / 8), blk, 0, stream>>>(qkv, hbuf,
                                                                   S);
  // 3) attn_out = hbuf @ Wo + bo
  gemm_wmma_f32_kernel<<<dim3(S / 256, D / 64), blk, 0, stream>>>(
      hbuf, Wo, bo, attn, S, D, D, D, D, D, 0);
  // 4) x1 = LN(x + attn_out)
  resid_ln_kernel<<<dim3(S), blk, 0, stream>>>(x, attn, ln1w, ln1b, x1);
  // 5) ff1 = gelu(x1 @ W1 + b1)
  gemm_wmma_f32_kernel<<<dim3(S / 256, HID / 64), blk, 0, stream>>>(
      x1, W1, b1, ff1, S, HID, D, D, HID, HID, 1);
  // 6) ff2 = ff1 @ W2 + b2
  gemm_wmma_f32_kernel<<<dim3(S / 256, D / 64), blk, 0, stream>>>(
      ff1, W2, b2, ff2, S, D, HID, HID, D, D, 0);
  // 7) out = LN(x1 + ff2)
  resid_ln_kernel<<<dim3(S), blk, 0, stream>>>(x1, ff2, ln2w, ln2b, out);
}